// MemoryNet_23502061044266
// MI455X (gfx1250) — compile-verified
//
#include <hip/hip_runtime.h>

// ---------------------------------------------------------------------------
// MI455X (gfx1250) implementation.
//
// Roofline: dominant cost is 50 sequential GEMVs over a 256 MB fp32 matrix
// (memory-bound, AI = 0.5 FLOP/B) plus a 512 MB streaming rank-1 update.
// Strategy: VALU fp32 FMA streaming with an L2-residency partition of M
// (160 MB regular-temporal resident + 96 MB nontemporal streaming), NT
// loads/stores on the Hebbian update, and V_WMMA_F32_16X16X4_F32 for the
// tiny batch-1 MLP front end (the only matrix-shaped math in the net).
// ---------------------------------------------------------------------------

typedef float v2f __attribute__((ext_vector_type(2)));
typedef float v8f __attribute__((ext_vector_type(8)));
typedef float f4  __attribute__((ext_vector_type(4)));

#define P        8192
#define NP       64            // p-chunks for the matvec partial reduction
#define PCHUNK   (P / NP)      // 128 rows per chunk
#define RES_ROWS 5120          // 160 MB of M kept L2-resident (192 MB L2)

#define KAPPA 0.8f
#define LAMDA 0.9f
#define YITA  0.1f

// d_out flat offsets (floats), return order: x_inf, g_, x_, p0, p_, M_new
#define OUT_XINF 0
#define OUT_G    100
#define OUT_X    164
#define OUT_P0   292
#define OUT_PV   8484
#define OUT_M    16676

__device__ __forceinline__ float f_p(float v) {
    v = (v >= 0.f) ? v : 0.01f * v;          // LeakyReLU(0.01)
    return fminf(fmaxf(v, -1.f), 1.f);       // clamp [-1, 1]
}

// ---------------------------------------------------------------------------
// One wave computes y[n0..n0+15] = v[0..K-1] @ W[K x N] + bias via
// V_WMMA_F32_16X16X4_F32, accumulating K in steps of 4. Row 0 of the A tile
// carries v (rows 1..15 zero), so D VGPR0 lanes 0..15 hold the result row.
// A layout (16x4 f32): lane<16 -> {K=k0,k0+1}, lane>=16 -> {K=k0+2,k0+3}.
// B layout (4x16 f32): VGPR0 = rows {k0, k0+2}, VGPR1 = rows {k0+1, k0+3},
// N = lane&15 across each half-wave.
//
// A-fill is branchless: every lane loads v[k0+kk]/v[k0+kk+1] (valid address
// for all lanes, broadcast within each half-wave) and non-M-row-0 lanes are
// zeroed with a select -> pure v_cndmask, no EXEC manipulation around the
// WMMA. Two independent accumulator chains hide the WMMA->WMMA RAW hazard.
// ---------------------------------------------------------------------------
__device__ __forceinline__ void gemv16_wmma(const float* __restrict__ v,
                                            const float* __restrict__ W,
                                            int N, int K, int n0,
                                            const float* __restrict__ bias,
                                            float* out, bool relu) {
    const int  lane  = threadIdx.x & 31;
    const int  kk    = (lane < 16) ? 0 : 2;
    const int  n     = lane & 15;
    const bool row0  = (n == 0);             // lanes 0 and 16 carry A row M=0
    v8f acc0 = {}, acc1 = {};
    for (int k0 = 0; k0 < K; k0 += 8) {
        // ---- K slice [k0, k0+4) -> acc0
        {
            const float ax = v[k0 + kk];
            const float ay = v[k0 + kk + 1];
            v2f a, b;
            a.x = row0 ? ax : 0.f;
            a.y = row0 ? ay : 0.f;
            b.x = W[(k0 + kk)     * N + n0 + n];
            b.y = W[(k0 + kk + 1) * N + n0 + n];
            acc0 = __builtin_amdgcn_wmma_f32_16x16x4_f32(false, a, false, b,
                                                         (short)0, acc0, false, false);
        }
        // ---- K slice [k0+4, k0+8) -> acc1 (independent chain)
        {
            const int k1 = k0 + 4;
            const float ax = v[k1 + kk];
            const float ay = v[k1 + kk + 1];
            v2f a, b;
            a.x = row0 ? ax : 0.f;
            a.y = row0 ? ay : 0.f;
            b.x = W[(k1 + kk)     * N + n0 + n];
            b.y = W[(k1 + kk + 1) * N + n0 + n];
            acc1 = __builtin_amdgcn_wmma_f32_16x16x4_f32(false, a, false, b,
                                                         (short)0, acc1, false, false);
        }
    }
    const v8f acc = acc0 + acc1;
    if (lane < 16) {
        float y = acc[0] + bias[n0 + lane];
        if (relu) y = fmaxf(y, 0.f);
        out[n0 + lane] = y;
    }
}

// ---------------------------------------------------------------------------
// Front end: both MLPs (WMMA), p0 = outer(x_, g_), h0 = f_p(tile(g_)).
// One block, 1024 threads = 32 waves.
// ---------------------------------------------------------------------------
__global__ void setup_kernel(const float* __restrict__ x,  const float* __restrict__ g,
                             const float* __restrict__ sw1, const float* __restrict__ sb1,
                             const float* __restrict__ sw2, const float* __restrict__ sb2,
                             const float* __restrict__ gw1, const float* __restrict__ gb1,
                             const float* __restrict__ gw2, const float* __restrict__ gb2,
                             float* __restrict__ dout, float* __restrict__ h0) {
    __shared__ float s_h[512];   // relu(x @ sw1 + sb1)
    __shared__ float g_h[256];   // relu(g @ gw1 + gb1)
    __shared__ float x_s[128];   // x_
    __shared__ float g_s[64];    // g_
    const int wid = threadIdx.x >> 5;

    // Stage 1: hidden layers (48 output tiles of 16 across 32 waves)
    for (int job = wid; job < 48; job += 32) {
        if (job < 32) gemv16_wmma(x, sw1, 512, 256, job * 16,        sb1, s_h, true);
        else          gemv16_wmma(g, gw1, 256, 128, (job - 32) * 16, gb1, g_h, true);
    }
    __syncthreads();

    // Stage 2: output layers (8 + 4 tiles)
    for (int job = wid; job < 12; job += 32) {
        if (job < 8) gemv16_wmma(s_h, sw2, 128, 512, job * 16,       sb2, x_s, false);
        else         gemv16_wmma(g_h, gw2, 64,  256, (job - 8) * 16, gb2, g_s, false);
    }
    __syncthreads();

    // Stage 3: emit g_, x_, p0; initialize h0 = f_p(query)
    const int t = threadIdx.x;
    if (t < 128) dout[OUT_X + t] = x_s[t];
    if (t < 64)  dout[OUT_G + t] = g_s[t];
    for (int i = t; i < P; i += 1024) {
        const int ix = i >> 6, ig = i & 63;
        dout[OUT_P0 + i] = x_s[ix] * g_s[ig];
        h0[i] = f_p(g_s[ig]);
    }
}

// ---------------------------------------------------------------------------
// Hop phase A: partial[chunk][q] = sum_{p in chunk} h[p] * M[p][q].
// Grid (P/1024, NP) x 256 threads; one float4 column strip per thread.
// Rows < RES_ROWS use RT loads (stay in 192 MB L2 across all 50 iters);
// the rest stream with nontemporal loads to avoid evicting the resident set.
// ---------------------------------------------------------------------------
__global__ void hop_matvec(const float* __restrict__ M,
                           const float* __restrict__ h,
                           float* __restrict__ partial) {
    const int    q4    = blockIdx.x * 256 + threadIdx.x;  // float4 column index
    const int    pbase = blockIdx.y * PCHUNK;
    const f4* __restrict__ M4 = (const f4*)M;
    f4 acc = {0.f, 0.f, 0.f, 0.f};
    if (pbase < RES_ROWS) {
#pragma unroll 4
        for (int p = pbase; p < pbase + PCHUNK; ++p) {
            const float hv = h[p];                         // uniform -> s_load
            const f4 m = M4[(size_t)p * (P / 4) + q4];
            acc += hv * m;
        }
    } else {
#pragma unroll 4
        for (int p = pbase; p < pbase + PCHUNK; ++p) {
            const float hv = h[p];
            const f4 m = __builtin_nontemporal_load(&M4[(size_t)p * (P / 4) + q4]);
            acc += hv * m;
        }
    }
    ((f4*)partial)[(size_t)blockIdx.y * (P / 4) + q4] = acc;
}

// Hop phase B: deterministic reduce of NP partials + pointwise attractor step.
__global__ void hop_update(const float* __restrict__ partial,
                           const float* __restrict__ hin,
                           float* __restrict__ hout) {
    const int q = blockIdx.x * blockDim.x + threadIdx.x;
    float mv = 0.f;
#pragma unroll 8
    for (int k = 0; k < NP; ++k) mv += partial[(size_t)k * P + q];
    const float hv = hin[q];
    hout[q] = f_p(hv * (KAPPA + mv));   // kappa*h + h*(h@M), then f_p
}

// p_ writeback, pd = p0 - h, x_out row-sums, classifier head.
__global__ void finalize_kernel(const float* __restrict__ h,
                                const float* __restrict__ pw,
                                const float* __restrict__ pb,
                                float* __restrict__ dout,
                                float* __restrict__ pd) {
    __shared__ float xo[128];
    const int t = threadIdx.x;
    const float* p0v = dout + OUT_P0;
    for (int i = t; i < P; i += 256) {
        const float hv = h[i];
        dout[OUT_PV + i] = hv;
        pd[i] = p0v[i] - hv;
    }
    if (t < 128) {
        float s = 0.f;
        for (int j = 0; j < 64; ++j) s += h[t * 64 + j];
        xo[t] = s;
    }
    __syncthreads();
    if (t < 100) {
        float s = pb[t];
        for (int i = 0; i < 128; ++i) s += xo[i] * pw[i * 100 + t];
        dout[OUT_XINF + t] = s;
    }
}

// Hebbian write: M_new[i][j] = 0.9*M[i][j] + 0.1*(p0_i + h_i)*(p0_j - h_j).
// Pure streaming: NT load of M (last use), NT store of M_new (never re-read).
__global__ void heb_update(const float* __restrict__ M,
                           const float* __restrict__ h,
                           const float* __restrict__ pd,
                           float* __restrict__ dout) {
    const int i = blockIdx.x;
    const float* p0v = dout + OUT_P0;
    const float a = YITA * (p0v[i] + h[i]);
    const f4* __restrict__ Mr = (const f4*)(M + (size_t)i * P);
    const f4* __restrict__ D  = (const f4*)pd;
    f4* __restrict__ O        = (f4*)(dout + OUT_M + (size_t)i * P);
#pragma unroll 4
    for (int j = threadIdx.x; j < P / 4; j += 256) {
        const f4 m = __builtin_nontemporal_load(&Mr[j]);
        const f4 d = D[j];
        const f4 o = LAMDA * m + a * d;
        __builtin_nontemporal_store(o, &O[j]);
    }
}

extern "C" void kernel_launch(void* const* d_in, const int* in_sizes, int n_in,
                              void* d_out, int out_size, void* d_ws, size_t ws_size,
                              hipStream_t stream) {
    const float* x   = (const float*)d_in[0];
    const float* g   = (const float*)d_in[1];
    const float* M   = (const float*)d_in[2];
    const float* sw1 = (const float*)d_in[3];
    const float* sb1 = (const float*)d_in[4];
    const float* sw2 = (const float*)d_in[5];
    const float* sb2 = (const float*)d_in[6];
    const float* gw1 = (const float*)d_in[7];
    const float* gb1 = (const float*)d_in[8];
    const float* gw2 = (const float*)d_in[9];
    const float* gb2 = (const float*)d_in[10];
    const float* pw  = (const float*)d_in[11];
    const float* pb  = (const float*)d_in[12];

    float* dout = (float*)d_out;
    float* ws   = (float*)d_ws;
    float* hA      = ws;               // 8192
    float* hB      = ws + P;           // 8192
    float* pd      = ws + 2 * P;       // 8192 (p0 - h)
    float* partial = ws + 3 * P;       // NP * 8192

    setup_kernel<<<1, 1024, 0, stream>>>(x, g, sw1, sb1, sw2, sb2,
                                         gw1, gb1, gw2, gb2, dout, hA);

    for (int it = 0; it < 50; ++it) {
        const float* hin  = (it & 1) ? hB : hA;
        float*       hout = (it & 1) ? hA : hB;
        hop_matvec<<<dim3(P / 1024, NP), 256, 0, stream>>>(M, hin, partial);
        hop_update<<<P / 256, 256, 0, stream>>>(partial, hin, hout);
    }
    // 50 steps (even) -> final h lands back in hA

    finalize_kernel<<<1, 256, 0, stream>>>(hA, pw, pb, dout, pd);
    heb_update<<<P, 256, 0, stream>>>(M, hA, pd, dout);
}